// RecommendationLoss_72919954752078
// MI455X (gfx1250) — compile-verified
//
#include <hip/hip_runtime.h>
#include <hip/hip_bf16.h>
#include <stdint.h>

typedef float v4f __attribute__((ext_vector_type(4)));

#define RL_B 8192
#define RL_L 1024
#define RL_MARGIN 0.1f
#define RL_LN2 0.6931471805599453094f
#define WAVES_PER_BLOCK 8

// Hardware log2 on the TRANS pipe (v_log_f32), co-issues with VALU.
__device__ __forceinline__ float fast_ln(float x) {
    return __builtin_amdgcn_logf(x) * RL_LN2;
}

// One wave (32 lanes) processes one row of 1024 elements.
// Scores are staged into LDS via the CDNA5 async-tensor path
// (global_load_async_to_lds_b128 + s_wait_asynccnt) because the hinge term
// needs a second pass over scores after the wave-wide "chosen" reduction.
__global__ __launch_bounds__(256) void rl_row_kernel(
    const float* __restrict__ scores,
    const int*   __restrict__ lens,
    const float* __restrict__ labels,
    float* __restrict__ bce_per,
    float* __restrict__ hinge_per,
    float* __restrict__ validf)
{
    __shared__ float smem[WAVES_PER_BLOCK * RL_L];   // 32 KB of the 320 KB WGP LDS

    const int lane = threadIdx.x & 31;
    const int wv   = threadIdx.x >> 5;
    const int row  = blockIdx.x * WAVES_PER_BLOCK + wv;

    const float* srow = scores + (size_t)row * RL_L;
    const float* lrow = labels + (size_t)row * RL_L;
    const int    len  = lens[row];

    // Low 32 bits of a generic LDS pointer are the LDS byte offset
    // (flat aperture lives in addr[63:32]); async-to-LDS adds LDS_BASE itself.
    const uint32_t lds_base = (uint32_t)(uintptr_t)&smem[wv * RL_L];

    // ---- stage this row's scores (4 KB) into LDS: 8 x b128 per lane ----
#pragma unroll
    for (int it = 0; it < 8; ++it) {
        uint32_t laddr = lds_base + (uint32_t)(it * 512 + lane * 16);
        uint64_t gaddr = (uint64_t)(uintptr_t)(srow + it * 128 + lane * 4);
        asm volatile("global_load_async_to_lds_b128 %0, %1, off"
                     :: "v"(laddr), "v"(gaddr) : "memory");
    }
    asm volatile("s_wait_asynccnt 0x0" ::: "memory");

    const v4f* lds_row = (const v4f*)&smem[wv * RL_L];

    // ---- pass 1: BCE + positive/negative bookkeeping ----
    float bce = 0.0f, pos_cnt = 0.0f, chosen_acc = 0.0f, neg_cnt = 0.0f;
    uint32_t negbits = 0u;

#pragma unroll
    for (int it = 0; it < 8; ++it) {
        const int c0 = it * 128 + lane * 4;
        v4f s = lds_row[it * 32 + lane];                                  // ds_load_b128
        v4f l = __builtin_nontemporal_load((const v4f*)(lrow + c0));      // TH=NT stream
#pragma unroll
        for (int k = 0; k < 4; ++k) {
            const bool inr = (c0 + k) < len;
            const float sv = s[k];
            const float lv = l[k];
            const float lp   = fmaxf(fast_ln(sv),          -100.0f);  // BCELoss log clamp
            const float l1mp = fmaxf(fast_ln(1.0f - sv),   -100.0f);  // log1p(-s), s<=1-1e-4
            const float bel  = -(lv * lp + (1.0f - lv) * l1mp);
            if (inr) {
                bce += bel;
                if (lv == 1.0f) { pos_cnt += 1.0f; chosen_acc += sv; }
                if (lv == 0.0f) { neg_cnt += 1.0f; negbits |= (1u << (it * 4 + k)); }
            }
        }
    }

    // ---- wave32 butterfly reductions ----
#pragma unroll
    for (int off = 16; off >= 1; off >>= 1) {
        bce        += __shfl_xor(bce,        off, 32);
        pos_cnt    += __shfl_xor(pos_cnt,    off, 32);
        chosen_acc += __shfl_xor(chosen_acc, off, 32);
        neg_cnt    += __shfl_xor(neg_cnt,    off, 32);
    }
    const float chosen = (pos_cnt > 0.0f) ? chosen_acc : -RL_MARGIN;

    // ---- pass 2: hinge over negatives (scores re-read from LDS) ----
    float hinge = 0.0f;
#pragma unroll
    for (int it = 0; it < 8; ++it) {
        v4f s = lds_row[it * 32 + lane];
#pragma unroll
        for (int k = 0; k < 4; ++k) {
            if (negbits & (1u << (it * 4 + k)))
                hinge += fmaxf(RL_MARGIN + s[k] - chosen, 0.0f);
        }
    }
#pragma unroll
    for (int off = 16; off >= 1; off >>= 1)
        hinge += __shfl_xor(hinge, off, 32);

    if (lane == 0) {
        // bce_per = (bce_el*mask).mean(axis=1) / mask.sum(axis=1) = sum/(L*len)
        const float bp    = bce / ((float)RL_L * (float)len);
        const bool  valid = (len > 0) && (neg_cnt > 0.0f);
        const float hp    = valid ? (hinge / fmaxf(neg_cnt, 1.0f)) : 0.0f;
        bce_per[row]   = bp;
        hinge_per[row] = hp;
        validf[row]    = valid ? 1.0f : 0.0f;
    }
}

// Deterministic final reduction over 8192 per-row values + similarity mean.
__global__ __launch_bounds__(1024) void rl_final_kernel(
    const float* __restrict__ bce_per,
    const float* __restrict__ hinge_per,
    const float* __restrict__ validf,
    const float* __restrict__ sim,
    float* __restrict__ out)
{
    __shared__ float red[1024];
    const int t = threadIdx.x;

    float a_b = 0.0f, a_h = 0.0f, a_v = 0.0f, a_s = 0.0f;
    for (int r = t; r < RL_B; r += 1024) {
        a_b += bce_per[r];
        a_h += hinge_per[r];
        a_v += validf[r];
        a_s += sim[r];
    }

    float vals[4] = { a_b, a_h, a_v, a_s };
    float totals[4];
#pragma unroll
    for (int q = 0; q < 4; ++q) {
        red[t] = vals[q];
        __syncthreads();
        for (int off = 512; off >= 1; off >>= 1) {
            if (t < off) red[t] += red[t + off];
            __syncthreads();
        }
        totals[q] = red[0];
        __syncthreads();
    }

    if (t == 0) {
        const float bce_loss   = totals[0] / (float)RL_B;
        const float vcnt       = totals[2];
        const float hinge_loss = (vcnt > 0.0f) ? (totals[1] / fmaxf(vcnt, 1.0f)) : 0.0f;
        const float sim_loss   = -totals[3] / (float)RL_B;
        const float combined   = hinge_loss + 1.0f * bce_loss + 1.0f * sim_loss;
        out[0] = combined;
        out[1] = hinge_loss;
        out[2] = bce_loss;
        out[3] = sim_loss;
    }
}

extern "C" void kernel_launch(void* const* d_in, const int* in_sizes, int n_in,
                              void* d_out, int out_size, void* d_ws, size_t ws_size,
                              hipStream_t stream) {
    const float* scores = (const float*)d_in[0];
    const int*   lens   = (const int*)d_in[1];
    const float* labels = (const float*)d_in[2];
    const float* sim    = (const float*)d_in[3];

    float* ws        = (float*)d_ws;
    float* bce_per   = ws;
    float* hinge_per = ws + RL_B;
    float* validf    = ws + 2 * RL_B;

    rl_row_kernel<<<RL_B / WAVES_PER_BLOCK, 256, 0, stream>>>(
        scores, lens, labels, bce_per, hinge_per, validf);
    rl_final_kernel<<<1, 1024, 0, stream>>>(
        bce_per, hinge_per, validf, sim, (float*)d_out);
}